// L1OutUB_13176959664365
// MI455X (gfx1250) — compile-verified
//
#include <hip/hip_runtime.h>
#include <hip/hip_bf16.h>
#include <math.h>

// Problem dims (from reference)
#define XD   128
#define YD   64
#define H2   256
#define BSZ  1024

typedef float v2f __attribute__((ext_vector_type(2)));
typedef float v8f __attribute__((ext_vector_type(8)));

// ---------------- Layer 1: H = relu(X @ W + b), X:[1024x128], W:[128x256] ----
// One 16x16 output tile per wave32, fp32 WMMA 16x16x4, K fully unrolled (32 wmma).
__global__ __launch_bounds__(256) void mlp_layer1(
    const float* __restrict__ X, const float* __restrict__ W,
    const float* __restrict__ bias, float* __restrict__ H)
{
    const int wid    = blockIdx.x * (blockDim.x >> 5) + (threadIdx.x >> 5);
    const int lane   = threadIdx.x & 31;
    const int half   = lane >> 4;     // 0: K even pair base, 1: K odd pair base
    const int l16    = lane & 15;
    const int tile_m = (wid >> 4) << 4;   // 64 M-tiles
    const int tile_n = (wid & 15) << 4;   // 16 N-tiles

    v8f acc = {};
    const float* xrow = X + (tile_m + l16) * XD;
    const int ncol = tile_n + l16;

#pragma unroll
    for (int k0 = 0; k0 < XD; k0 += 4) {
        const int k = k0 + 2 * half;
        v2f a; a.x = xrow[k];                 a.y = xrow[k + 1];
        v2f b; b.x = W[(k + 0) * H2 + ncol];  b.y = W[(k + 1) * H2 + ncol];
        acc = __builtin_amdgcn_wmma_f32_16x16x4_f32(
            false, a, false, b, (short)0, acc, false, false);
    }

    const float bv = bias[ncol];
#pragma unroll
    for (int v = 0; v < 8; ++v) {
        const int m = tile_m + v + 8 * half;
        float r = acc[v] + bv;
        H[m * H2 + ncol] = r > 0.f ? r : 0.f;
    }
}

// ---------------- Layer 2: O = (H @ W + b) [optional tanh], H:[1024x256], W:[256x64]
__global__ __launch_bounds__(256) void mlp_layer2(
    const float* __restrict__ H, const float* __restrict__ W,
    const float* __restrict__ bias, float* __restrict__ O, int apply_tanh)
{
    const int wid    = blockIdx.x * (blockDim.x >> 5) + (threadIdx.x >> 5);
    const int lane   = threadIdx.x & 31;
    const int half   = lane >> 4;
    const int l16    = lane & 15;
    const int tile_m = (wid >> 2) << 4;   // 64 M-tiles
    const int tile_n = (wid & 3) << 4;    // 4  N-tiles

    v8f acc = {};
    const float* hrow = H + (tile_m + l16) * H2;
    const int ncol = tile_n + l16;

#pragma unroll
    for (int k0 = 0; k0 < H2; k0 += 4) {
        const int k = k0 + 2 * half;
        v2f a; a.x = hrow[k];                 a.y = hrow[k + 1];
        v2f b; b.x = W[(k + 0) * YD + ncol];  b.y = W[(k + 1) * YD + ncol];
        acc = __builtin_amdgcn_wmma_f32_16x16x4_f32(
            false, a, false, b, (short)0, acc, false, false);
    }

    const float bv = bias[ncol];
#pragma unroll
    for (int v = 0; v < 8; ++v) {
        const int m = tile_m + v + 8 * half;
        float r = acc[v] + bv;
        if (apply_tanh) r = tanhf(r);
        O[m * YD + ncol] = r;
    }
}

// ---------------- Zero the accumulator slab (258 floats) ----------------
__global__ void zero_accum(float* __restrict__ acc, int n)
{
    int i = blockIdx.x * blockDim.x + threadIdx.x;
    if (i < n) acc[i] = 0.f;
}

// ---------------- Per-row reductions ----------------
// 64 blocks x 256 threads; block handles 16 rows. Produces:
//   Sinv[d]  = sum_i inv[i,d],  Smuinv[d] = sum_i mu*inv,  Sy[d], Sy2[d]
//   Spos     = sum_i positive[i],  SC = sum_i C_i
__global__ __launch_bounds__(256) void row_reduce(
    const float* __restrict__ MU, const float* __restrict__ LV,
    const float* __restrict__ Y,
    float* __restrict__ Sinv, float* __restrict__ Smuinv,
    float* __restrict__ Sy,   float* __restrict__ Sy2,
    float* __restrict__ Spos, float* __restrict__ SC)
{
    const int t  = threadIdx.x;
    const int d  = t & 63;
    const int rg = t >> 6;   // 0..3 row groups

    float inv_s = 0.f, muinv_s = 0.f, y_s = 0.f, y2_s = 0.f, pos_s = 0.f, c_s = 0.f;
    for (int rr = rg; rr < 16; rr += 4) {
        const int i = blockIdx.x * 16 + rr;
        const float lv  = LV[i * YD + d];
        const float inv = 0.5f * __expf(-lv);     // 1 / (2 e^{lv})
        const float m   = MU[i * YD + d];
        const float yv  = Y [i * YD + d];
        const float dm  = m - yv;
        pos_s  += -dm * dm * inv - 0.5f * lv;
        c_s    += -m  * m  * inv - 0.5f * lv;
        inv_s  += inv;
        muinv_s+= m * inv;
        y_s    += yv;
        y2_s   += yv * yv;
    }

    __shared__ float sh[4][64][4];
    __shared__ float shp[256], shc[256];
    sh[rg][d][0] = inv_s; sh[rg][d][1] = muinv_s;
    sh[rg][d][2] = y_s;   sh[rg][d][3] = y2_s;
    shp[t] = pos_s; shc[t] = c_s;
    __syncthreads();

    if (rg == 0) {
        float a = 0.f, b = 0.f, c = 0.f, e = 0.f;
        for (int g = 0; g < 4; ++g) {
            a += sh[g][d][0]; b += sh[g][d][1];
            c += sh[g][d][2]; e += sh[g][d][3];
        }
        atomicAdd(&Sinv[d], a);  atomicAdd(&Smuinv[d], b);
        atomicAdd(&Sy[d], c);    atomicAdd(&Sy2[d], e);
    }
    for (int s = 128; s > 0; s >>= 1) {
        if (t < s) { shp[t] += shp[t + s]; shc[t] += shc[t + s]; }
        __syncthreads();
    }
    if (t == 0) { atomicAdd(Spos, shp[0]); atomicAdd(SC, shc[0]); }
}

// ---------------- Finalize: combine the rank-1 sums into the scalar ----------
__global__ void finalize(
    const float* __restrict__ Sinv, const float* __restrict__ Smuinv,
    const float* __restrict__ Sy,   const float* __restrict__ Sy2,
    const float* __restrict__ Spos, const float* __restrict__ SC,
    float* __restrict__ out)
{
    __shared__ float sh[64];
    const int d = threadIdx.x;
    sh[d] = -Sy2[d] * Sinv[d] + 2.f * Sy[d] * Smuinv[d];
    __syncthreads();
    for (int s = 32; s > 0; s >>= 1) {
        if (d < s) sh[d] += sh[d + s];
        __syncthreads();
    }
    if (d == 0) {
        const float Bf = (float)BSZ;
        const float all_sum  = sh[0] + Bf * SC[0];              // = sum_{i,j} all_probs
        const float lse_mask = logf((Bf - 1.f) + expf(-20.f));
        const float res = Spos[0] / Bf - all_sum / (Bf * Bf) - (lse_mask - logf(Bf - 1.f));
        out[0] = res;
    }
}

extern "C" void kernel_launch(void* const* d_in, const int* in_sizes, int n_in,
                              void* d_out, int out_size, void* d_ws, size_t ws_size,
                              hipStream_t stream) {
    const float* X   = (const float*)d_in[0];   // [1024,128]
    const float* Y   = (const float*)d_in[1];   // [1024,64]
    const float* Wm1 = (const float*)d_in[2];   // [128,256]
    const float* bm1 = (const float*)d_in[3];   // [256]
    const float* Wm2 = (const float*)d_in[4];   // [256,64]
    const float* bm2 = (const float*)d_in[5];   // [64]
    const float* Wv1 = (const float*)d_in[6];
    const float* bv1 = (const float*)d_in[7];
    const float* Wv2 = (const float*)d_in[8];
    const float* bv2 = (const float*)d_in[9];
    float* out = (float*)d_out;

    // Workspace layout (floats)
    float* ws   = (float*)d_ws;
    float* Hm   = ws;                       // 1024*256
    float* Hv   = Hm + BSZ * H2;            // 1024*256
    float* MU   = Hv + BSZ * H2;            // 1024*64
    float* LV   = MU + BSZ * YD;            // 1024*64
    float* Sinv   = LV + BSZ * YD;          // 64
    float* Smuinv = Sinv + 64;              // 64
    float* Sy     = Smuinv + 64;            // 64
    float* Sy2    = Sy + 64;                // 64
    float* Spos   = Sy2 + 64;               // 1
    float* SC     = Spos + 1;               // 1

    zero_accum<<<1, 256, 0, stream>>>(Sinv, 258);

    // Layer 1 (both heads): 1024 tiles, 8 waves/block -> 128 blocks
    mlp_layer1<<<128, 256, 0, stream>>>(X, Wm1, bm1, Hm);
    mlp_layer1<<<128, 256, 0, stream>>>(X, Wv1, bv1, Hv);

    // Layer 2 (both heads): 256 tiles, 8 waves/block -> 32 blocks
    mlp_layer2<<<32, 256, 0, stream>>>(Hm, Wm2, bm2, MU, 0);
    mlp_layer2<<<32, 256, 0, stream>>>(Hv, Wv2, bv2, LV, 1);

    // Rank-1 collapse of the BxB pairwise term + positive/row sums
    row_reduce<<<64, 256, 0, stream>>>(MU, LV, Y, Sinv, Smuinv, Sy, Sy2, Spos, SC);

    // Scalar epilogue
    finalize<<<1, 64, 0, stream>>>(Sinv, Smuinv, Sy, Sy2, Spos, SC, out);
}